// SimpleHubDiscriminator_55155970015930
// MI455X (gfx1250) — compile-verified
//
#include <hip/hip_runtime.h>

#define NN 100000      // nodes
#define NE 3200000     // edges
#define NG 256         // graphs
#define ND 7           // node feature dim
#define NH 64          // hidden
#define NL 3           // layers
#define EPSV 1e-5f

typedef __attribute__((ext_vector_type(2))) float v2f;
typedef __attribute__((ext_vector_type(8))) float v8f;

// ---------------------------------------------------------------- utilities
__global__ void fill_f32(float* __restrict__ p, float v, int n) {
    int t = blockIdx.x * 256 + threadIdx.x;
    if (t < n) p[t] = v;
}

// deg[dst] += 1 (deg pre-initialized to 1.0 for the self loop)
__global__ void deg_kernel(const int* __restrict__ dst, float* __restrict__ deg, int n) {
    int t = blockIdx.x * 256 + threadIdx.x;
    if (t < n) atomicAdd(&deg[dst[t]], 1.0f);
}

__global__ void dinv_kernel(const float* __restrict__ deg, float* __restrict__ dinv, int n) {
    int t = blockIdx.x * 256 + threadIdx.x;
    if (t < n) dinv[t] = rsqrtf(deg[t]);   // deg >= 1 always (self loop)
}

__global__ void cnt_kernel(const int* __restrict__ batch, float* __restrict__ cnt, int n) {
    int t = blockIdx.x * 256 + threadIdx.x;
    if (t < n) atomicAdd(&cnt[batch[t]], 1.0f);
}

// h = x @ Win + bin     (K=7, tiny; VALU kernel)
__global__ void input_proj(const float* __restrict__ x, const float* __restrict__ Win,
                           const float* __restrict__ binp, float* __restrict__ h, int n) {
    int t = blockIdx.x * 256 + threadIdx.x;
    if (t >= n) return;
    int v = t >> 6, j = t & 63;
    float acc = binp[j];
#pragma unroll
    for (int k = 0; k < ND; ++k) acc = fmaf(x[v * ND + k], Win[k * NH + j], acc);
    h[t] = acc;
}

// -------------------------------------------------- fp32 WMMA GEMM  [M,64]x[64,64]
// One wave -> one 16x16 output tile, 16 x V_WMMA_F32_16X16X4_F32 along K.
// Block: 256 threads = 8 waves = 2 row-tiles x 4 col-tiles (32 rows / block).
// M must be a multiple of 32 (100000 and 256 both are) -> EXEC stays all-ones.
__global__ __launch_bounds__(256) void gemm64_wmma(const float* __restrict__ A,
                                                   const float* __restrict__ W,
                                                   const float* __restrict__ bias,
                                                   float* __restrict__ out,
                                                   int M, int relu) {
    int wave = threadIdx.x >> 5;
    int lane = threadIdx.x & 31;
    int rowBase = (blockIdx.x * 2 + (wave >> 2)) * 16;
    int colBase = (wave & 3) * 16;
    if (rowBase >= M) return;              // never diverges within a wave
    int l  = lane & 15;
    int hi = lane >> 4;                    // lane-half selects K sub-pair

    v8f acc = {};
    const float* ap = A + (rowBase + l) * NH + hi * 2;        // A[m][k], m=rowBase+l
    const float* bp = W + (hi * 2) * NH + colBase + l;        // W[k][n], n=colBase+l
#pragma unroll
    for (int kk = 0; kk < 16; ++kk) {
        v2f a, b;
        a.x = ap[0];  a.y = ap[1];                            // K = kk*4 + hi*2 (+1)
        b.x = bp[0];  b.y = bp[NH];
        acc = __builtin_amdgcn_wmma_f32_16x16x4_f32(false, a, false, b,
                                                    (short)0, acc, false, false);
        ap += 4;
        bp += 4 * NH;
    }
    float bv = bias ? bias[colBase + l] : 0.0f;
#pragma unroll
    for (int r = 0; r < 8; ++r) {                             // C/D layout: VGPR r ->
        int m = rowBase + hi * 8 + r;                         //   M = hi*8 + r, N = l
        float v = acc[r] + bv;
        if (relu) v = fmaxf(v, 0.0f);
        out[m * NH + colBase + l] = v;
    }
}

// c = convB + self-loop term xw[v] * dinv[v]^2
__global__ void init_c(const float* __restrict__ xw, const float* __restrict__ dinv,
                       const float* __restrict__ b, float* __restrict__ c, int n) {
    int t = blockIdx.x * 256 + threadIdx.x;
    if (t >= n) return;
    int v = t >> 6, j = t & 63;
    float di = dinv[v];
    c[t] = fmaf(xw[t], di * di, b[j]);
}

// c[dst] += xw[src] * dinv[src]*dinv[dst]   (16 threads/edge, float4 gather, 4 atomics)
__global__ void edge_scatter(const int* __restrict__ src, const int* __restrict__ dst,
                             const float* __restrict__ dinv, const float* __restrict__ xw,
                             float* __restrict__ c, int nq) {
    int t = blockIdx.x * 256 + threadIdx.x;
    if (t >= nq) return;
    int e = t >> 4, q = t & 15;
    int s = src[e], d = dst[e];
    float nrm = dinv[s] * dinv[d];
    const float4* xs = reinterpret_cast<const float4*>(xw + s * NH);
    float4 v = xs[q];
    float* cd = c + d * NH + q * 4;
    atomicAdd(cd + 0, v.x * nrm);
    atomicAdd(cd + 1, v.y * nrm);
    atomicAdd(cd + 2, v.z * nrm);
    atomicAdd(cd + 3, v.w * nrm);
}

// ---------------------------------------------------------------- GraphNorm
__global__ void gnorm_sum(const float* __restrict__ c, const int* __restrict__ batch,
                          float* __restrict__ gsum, int n) {
    int t = blockIdx.x * 256 + threadIdx.x;
    if (t >= n) return;
    int v = t >> 6, j = t & 63;
    atomicAdd(&gsum[batch[v] * NH + j], c[t]);
}

__global__ void gnorm_center(float* __restrict__ c, const int* __restrict__ batch,
                             const float* __restrict__ gsum, const float* __restrict__ cnt,
                             const float* __restrict__ mscale, float* __restrict__ gsq, int n) {
    int t = blockIdx.x * 256 + threadIdx.x;
    if (t >= n) return;
    int v = t >> 6, j = t & 63;
    int g = batch[v];
    float cn = fmaxf(cnt[g], 1.0f);
    float mean = gsum[g * NH + j] / cn;
    float o = c[t] - mscale[j] * mean;
    c[t] = o;
    atomicAdd(&gsq[g * NH + j], o * o);
}

__global__ void gnorm_final(const float* __restrict__ c, const int* __restrict__ batch,
                            const float* __restrict__ gsq, const float* __restrict__ cnt,
                            const float* __restrict__ w, const float* __restrict__ b,
                            float* __restrict__ h, int n) {
    int t = blockIdx.x * 256 + threadIdx.x;
    if (t >= n) return;
    int v = t >> 6, j = t & 63;
    int g = batch[v];
    float cn = fmaxf(cnt[g], 1.0f);
    float inv = rsqrtf(gsq[g * NH + j] / cn + EPSV);
    float val = fmaf(w[j] * c[t], inv, b[j]);
    h[t] += fmaxf(val, 0.0f);               // relu then residual
}

// ---------------------------------------------------------------- pooling + head
__global__ void pool_sum(const float* __restrict__ h, const int* __restrict__ batch,
                         float* __restrict__ emb, int n) {
    int t = blockIdx.x * 256 + threadIdx.x;
    if (t >= n) return;
    int v = t >> 6, j = t & 63;
    atomicAdd(&emb[batch[v] * NH + j], h[t]);
}

__global__ void pool_div(float* __restrict__ emb, const float* __restrict__ cnt, int n) {
    int t = blockIdx.x * 256 + threadIdx.x;
    if (t >= n) return;
    emb[t] /= fmaxf(cnt[t >> 6], 1.0f);
}

__global__ void head(const float* __restrict__ z, const float* __restrict__ W2,
                     const float* __restrict__ B2, float* __restrict__ out, int n) {
    int t = blockIdx.x * 256 + threadIdx.x;
    if (t >= n) return;
    int g = t >> 1, o = t & 1;
    float acc = B2[o];
#pragma unroll
    for (int j = 0; j < NH; ++j) acc = fmaf(z[g * NH + j], W2[j * 2 + o], acc);
    out[t] = acc;
}

// ---------------------------------------------------------------- launcher
extern "C" void kernel_launch(void* const* d_in, const int* in_sizes, int n_in,
                              void* d_out, int out_size, void* d_ws, size_t ws_size,
                              hipStream_t stream) {
    const float* x     = (const float*)d_in[0];
    const int*   ei    = (const int*)d_in[1];
    const int*   batch = (const int*)d_in[2];
    const float* Win   = (const float*)d_in[3];
    const float* binp  = (const float*)d_in[4];
    const float* convW = (const float*)d_in[5];
    const float* convB = (const float*)d_in[6];
    const float* gnW   = (const float*)d_in[7];
    const float* gnB   = (const float*)d_in[8];
    const float* gnS   = (const float*)d_in[9];
    const float* W1    = (const float*)d_in[10];
    const float* B1    = (const float*)d_in[11];
    const float* W2    = (const float*)d_in[12];
    const float* B2    = (const float*)d_in[13];
    float* out = (float*)d_out;

    float* ws   = (float*)d_ws;
    float* deg  = ws;  ws += NN;
    float* dinv = ws;  ws += NN;
    float* h    = ws;  ws += (size_t)NN * NH;
    float* xw   = ws;  ws += (size_t)NN * NH;
    float* c    = ws;  ws += (size_t)NN * NH;
    float* cnt  = ws;  ws += NG;
    float* gsum = ws;  ws += NG * NH;
    float* gsq  = ws;  ws += NG * NH;
    float* emb  = ws;  ws += NG * NH;
    float* z    = ws;  ws += NG * NH;

    const int* src = ei;
    const int* dst = ei + NE;

    auto nb = [](int n) { return (n + 255) / 256; };

    // degrees (self-loop = 1.0 baseline), graph counts, input projection
    fill_f32<<<nb(NN), 256, 0, stream>>>(deg, 1.0f, NN);
    fill_f32<<<nb(NG), 256, 0, stream>>>(cnt, 0.0f, NG);
    deg_kernel<<<nb(NE), 256, 0, stream>>>(dst, deg, NE);
    dinv_kernel<<<nb(NN), 256, 0, stream>>>(deg, dinv, NN);
    cnt_kernel<<<nb(NN), 256, 0, stream>>>(batch, cnt, NN);
    input_proj<<<nb(NN * NH), 256, 0, stream>>>(x, Win, binp, h, NN * NH);

    for (int L = 0; L < NL; ++L) {
        gemm64_wmma<<<NN / 32, 256, 0, stream>>>(h, convW + L * NH * NH, nullptr, xw, NN, 0);
        init_c<<<nb(NN * NH), 256, 0, stream>>>(xw, dinv, convB + L * NH, c, NN * NH);
        edge_scatter<<<nb(NE * 16), 256, 0, stream>>>(src, dst, dinv, xw, c, NE * 16);
        fill_f32<<<nb(NG * NH), 256, 0, stream>>>(gsum, 0.0f, NG * NH);
        fill_f32<<<nb(NG * NH), 256, 0, stream>>>(gsq, 0.0f, NG * NH);
        gnorm_sum<<<nb(NN * NH), 256, 0, stream>>>(c, batch, gsum, NN * NH);
        gnorm_center<<<nb(NN * NH), 256, 0, stream>>>(c, batch, gsum, cnt, gnS + L * NH, gsq, NN * NH);
        gnorm_final<<<nb(NN * NH), 256, 0, stream>>>(c, batch, gsq, cnt, gnW + L * NH, gnB + L * NH, h, NN * NH);
    }

    fill_f32<<<nb(NG * NH), 256, 0, stream>>>(emb, 0.0f, NG * NH);
    pool_sum<<<nb(NN * NH), 256, 0, stream>>>(h, batch, emb, NN * NH);
    pool_div<<<nb(NG * NH), 256, 0, stream>>>(emb, cnt, NG * NH);
    gemm64_wmma<<<NG / 32, 256, 0, stream>>>(emb, W1, B1, z, NG, 1);
    head<<<nb(NG * 2), 256, 0, stream>>>(z, W2, B2, out, NG * 2);
}